// HumanPoseModule_32066225832144
// MI455X (gfx1250) — compile-verified
//
#include <hip/hip_runtime.h>
#include <math.h>

#define DEVFN __device__ __forceinline__

typedef unsigned int v4u __attribute__((ext_vector_type(4)));
typedef int          v8i __attribute__((ext_vector_type(8)));
typedef int          v4i __attribute__((ext_vector_type(4)));

struct M3 { float a[9]; };   // row-major 3x3

DEVFN float rcpf(float x)  { return __builtin_amdgcn_rcpf(x); }   // v_rcp_f32
DEVFN float rsqf(float x)  { return __builtin_amdgcn_rsqf(x); }   // v_rsq_f32
DEVFN float sqrf(float x)  { return __builtin_amdgcn_sqrtf(x); }  // v_sqrt_f32

// ---- TDM: 1-D contiguous DMA of `ndw` dwords, global -> LDS ---------------
// D# layout per CDNA5 ISA:
//  g0: [1:0]=count=1 | [63:32]=lds_addr | [120:64]=global_addr | [127:126]=type=2
//  g1: [17:16]=data_size(2=4B) | tensor_dim0[79:48]=N | tensor_dim1[111:80]=1
//      tile_dim0[127:112]=N (N<=65535) | tile_dim1[143:128]=1 | tile_dim2=0
//      tensor_dim0_stride[207:160]=N | tensor_dim1_stride[255:208]=N
//  remaining groups: zero (tensor rank <= 2)
DEVFN void tdm_load_1d(unsigned int lds_off, unsigned long long gaddr,
                       unsigned int ndw) {
    v4u g0;
    g0.x = 1u;                                        // count=1 (user D#)
    g0.y = lds_off;                                   // lds_addr (bytes)
    g0.z = (unsigned int)gaddr;                       // global_addr[31:0]
    g0.w = (unsigned int)(gaddr >> 32) | (2u << 30);  // ga[56:32] | type=2
    v8i g1;
    g1[0] = (int)(2u << 16);                          // data_size=4B, mask=0
    g1[1] = (int)((ndw & 0xFFFFu) << 16);             // tensor_dim0[15:0]
    g1[2] = (int)((ndw >> 16) | (1u << 16));          // td0[31:16] | tensor_dim1=1
    g1[3] = (int)((ndw & 0xFFFFu) << 16);             // tile_dim0 = ndw
    g1[4] = 1;                                        // tile_dim1=1, tile_dim2=0
    g1[5] = (int)ndw;                                 // stride0[31:0]
    g1[6] = (int)(((ndw & 0xFFFFu) << 16) | (ndw >> 16)); // s0[47:32]|s1[15:0]
    g1[7] = (int)(ndw >> 16);                         // stride1[47:16]
    v4i z4 = {0, 0, 0, 0};
    v8i z8 = {0, 0, 0, 0, 0, 0, 0, 0};
    __builtin_amdgcn_tensor_load_to_lds(g0, g1, z4, z4, z8, 0);
}

// ---- 6D -> rotation matrix (rows b1,b2,b3) --------------------------------
DEVFN M3 rot6d(const float* d) {
    float ax = d[0], ay = d[1], az = d[2];
    float bx = d[3], by = d[4], bz = d[5];
    float n1 = fmaf(ax, ax, fmaf(ay, ay, az * az));
    float r1 = rsqf(n1);
    float b1x = ax * r1, b1y = ay * r1, b1z = az * r1;
    float dp = fmaf(b1x, bx, fmaf(b1y, by, b1z * bz));
    float cx = fmaf(-dp, b1x, bx);
    float cy = fmaf(-dp, b1y, by);
    float cz = fmaf(-dp, b1z, bz);
    float n2 = fmaf(cx, cx, fmaf(cy, cy, cz * cz));
    float r2 = rsqf(n2);
    float b2x = cx * r2, b2y = cy * r2, b2z = cz * r2;
    float b3x = fmaf(b1y, b2z, -b1z * b2y);
    float b3y = fmaf(b1z, b2x, -b1x * b2z);
    float b3z = fmaf(b1x, b2y, -b1y * b2x);
    M3 m;
    m.a[0] = b1x; m.a[1] = b1y; m.a[2] = b1z;
    m.a[3] = b2x; m.a[4] = b2y; m.a[5] = b2z;
    m.a[6] = b3x; m.a[7] = b3y; m.a[8] = b3z;
    return m;
}

// C = A^T * B
DEVFN M3 m3mulTN(const M3& A, const M3& B) {
    M3 C;
#pragma unroll
    for (int i = 0; i < 3; ++i)
#pragma unroll
        for (int j = 0; j < 3; ++j) {
            float s = A.a[0 + i] * B.a[0 + j];
            s = fmaf(A.a[3 + i], B.a[3 + j], s);
            s = fmaf(A.a[6 + i], B.a[6 + j], s);
            C.a[i * 3 + j] = s;
        }
    return C;
}

// atan2(n, w) with n >= 0, result in [0, pi].  Minimax odd poly on [0,1].
DEVFN float atan2_pos(float n, float w) {
    float aw = fabsf(w);
    float mn = fminf(n, aw);
    float mx = fmaxf(n, aw);
    float t  = mn * rcpf(fmaxf(mx, 1e-37f));
    float s2 = t * t;
    float u;
    u =                 0.00282363896258175373f;
    u = fmaf(u, s2,   -0.01595690287649631500f);
    u = fmaf(u, s2,    0.04250498861074447632f);
    u = fmaf(u, s2,   -0.07489009201526641846f);
    u = fmaf(u, s2,    0.10634793341159820557f);
    u = fmaf(u, s2,   -0.14202736318111419678f);
    u = fmaf(u, s2,    0.19992695748805999756f);
    u = fmaf(u, s2,   -0.33333101868629455566f);
    float r = fmaf(t * s2, u, t);
    r = (n > aw)   ? (1.57079632679489661923f - r) : r;
    r = (w < 0.0f) ? (3.14159265358979323846f - r) : r;
    return r;
}

// matrix -> quaternion -> axis-angle
DEVFN void mat_to_aa(const M3& R, float* o) {
    float m00 = R.a[0], m01 = R.a[1], m02 = R.a[2];
    float m10 = R.a[3], m11 = R.a[4], m12 = R.a[5];
    float m20 = R.a[6], m21 = R.a[7], m22 = R.a[8];

    float t0 = 1.0f + m00 + m11 + m22;
    float t1 = 1.0f + m00 - m11 - m22;
    float t2 = 1.0f - m00 + m11 - m22;
    float t3 = 1.0f - m00 - m11 + m22;

    int   idx  = 0;
    float best = t0;
    if (t1 > best) { best = t1; idx = 1; }
    if (t2 > best) { best = t2; idx = 2; }
    if (t3 > best) { best = t3; idx = 3; }
    float diag = fmaxf(best, 0.0f);
    float qa   = sqrf(diag);

    float A = m21 - m12, Bv = m02 - m20, C = m10 - m01;
    float D = m10 + m01, E  = m02 + m20, F = m12 + m21;

    float w = (idx == 0) ? diag : (idx == 1) ? A    : (idx == 2) ? Bv   : C;
    float x = (idx == 0) ? A    : (idx == 1) ? diag : (idx == 2) ? D    : E;
    float y = (idx == 0) ? Bv   : (idx == 1) ? D    : (idx == 2) ? diag : F;
    float z = (idx == 0) ? C    : (idx == 1) ? E    : (idx == 2) ? F    : diag;

    float inv = 0.5f * rcpf(fmaxf(qa, 0.1f));

    float nr   = sqrf(fmaf(x, x, fmaf(y, y, z * z)));
    float half = atan2_pos(nr, w);
    float ang  = 2.0f * half;
    bool  sml  = ang < 1e-6f;
    float sho  = sml ? fmaf(ang * ang, -(1.0f / 48.0f), 0.5f)
                     : __sinf(half) * rcpf(ang);
    float s    = inv * rcpf(sho);
    o[0] = x * s; o[1] = y * s; o[2] = z * s;
}

// ---------------------------------------------------------------------------
// 128 poses per block.  Wave 0 DMA-streams the block's contiguous inputs
// into LDS via the Tensor Data Mover, waits TENSORcnt, then all lanes read
// their pose from LDS and compute.  local[j] = R_par^T R_j (root cancels).
// ---------------------------------------------------------------------------
#define POSES_PER_BLOCK 128

__global__ __launch_bounds__(POSES_PER_BLOCK)
void pose_aa_kernel(const float* __restrict__ glb,
                    const float* __restrict__ ori,
                    float* __restrict__ out, int npose)
{
    __shared__ float sm[POSES_PER_BLOCK * 60 + POSES_PER_BLOCK * 36]; // 48 KB

    const int lane = threadIdx.x;
    const int base = blockIdx.x * POSES_PER_BLOCK;

    // one DMA issue per instruction per wave -> only wave 0 issues
    if ((threadIdx.x >> 5) == 0) {
        unsigned int npt = (unsigned int)((npose - base < POSES_PER_BLOCK)
                                              ? (npose - base) : POSES_PER_BLOCK);
        unsigned int lds0 = (unsigned int)(unsigned long long)(void*)&sm[0];
        tdm_load_1d(lds0,
                    (unsigned long long)(const void*)(glb + (size_t)base * 60),
                    npt * 60u);
        tdm_load_1d(lds0 + POSES_PER_BLOCK * 60u * 4u,
                    (unsigned long long)(const void*)(ori + (size_t)base * 36),
                    npt * 36u);
        __builtin_amdgcn_s_wait_tensorcnt(0);
    }
    __syncthreads();

    int t = base + lane;
    if (t >= npose) return;

    // per-pose copies out of LDS (aligned b128 reads: 240B / 144B strides)
    float gb[60];
    {
        const float4* p = (const float4*)(&sm[lane * 60]);
#pragma unroll
        for (int i = 0; i < 15; ++i) {
            float4 v = p[i];
            gb[4 * i + 0] = v.x; gb[4 * i + 1] = v.y;
            gb[4 * i + 2] = v.z; gb[4 * i + 3] = v.w;
        }
    }
    float ob[36];
    {
        const float4* p = (const float4*)(&sm[POSES_PER_BLOCK * 60 + lane * 36]);
#pragma unroll
        for (int i = 0; i < 9; ++i) {
            float4 v = p[i];
            ob[4 * i + 0] = v.x; ob[4 * i + 1] = v.y;
            ob[4 * i + 2] = v.z; ob[4 * i + 3] = v.w;
        }
    }

    float outv[72];
#pragma unroll
    for (int k = 0; k < 3; ++k) {
        outv[7 * 3 + k] = 0.0f;  outv[8 * 3 + k] = 0.0f;
        outv[10 * 3 + k] = 0.0f; outv[11 * 3 + k] = 0.0f;
        outv[20 * 3 + k] = 0.0f; outv[21 * 3 + k] = 0.0f;
        outv[22 * 3 + k] = 0.0f; outv[23 * 3 + k] = 0.0f;
    }

    M3 R[24];
    R[0] = rot6d(&ob[0]);
    mat_to_aa(R[0], &outv[0]);

#define LEAF(J, SRCBUF, OFF)                                   \
    {                                                          \
        R[J] = rot6d(&SRCBUF[(OFF) * 6]);                      \
        mat_to_aa(R[J], &outv[(J) * 3]);                       \
    }
#define STEP(J, PARJ, SRCBUF, OFF)                             \
    {                                                          \
        R[J] = rot6d(&SRCBUF[(OFF) * 6]);                      \
        M3 L = m3mulTN(R[PARJ], R[J]);                         \
        mat_to_aa(L, &outv[(J) * 3]);                          \
    }

    LEAF(1,      gb, 0)
    STEP(4,  1,  ob, 1)
    LEAF(2,      gb, 1)
    STEP(5,  2,  ob, 2)
    LEAF(3,      gb, 2)
    STEP(6,  3,  gb, 3)
    STEP(9,  6,  gb, 4)
    STEP(12, 9,  gb, 5)
    STEP(15, 12, ob, 3)
    STEP(13, 9,  gb, 6)
    STEP(16, 13, gb, 8)
    STEP(18, 16, ob, 4)
    STEP(14, 9,  gb, 7)
    STEP(17, 14, gb, 9)
    STEP(19, 17, ob, 5)
#undef LEAF
#undef STEP

    float4* po = (float4*)(out + (size_t)t * 72);
#pragma unroll
    for (int i = 0; i < 18; ++i) {
        po[i] = make_float4(outv[4 * i + 0], outv[4 * i + 1],
                            outv[4 * i + 2], outv[4 * i + 3]);
    }
}

extern "C" void kernel_launch(void* const* d_in, const int* in_sizes, int n_in,
                              void* d_out, int out_size, void* d_ws, size_t ws_size,
                              hipStream_t stream)
{
    const float* glb = (const float*)d_in[0];   // (B,T,10,6) f32
    const float* ori = (const float*)d_in[1];   // (B,T,6,6)  f32
    float* out = (float*)d_out;                 // (B,T,24,3) f32

    int npose = in_sizes[0] / 60;               // B*T
    int grid  = (npose + POSES_PER_BLOCK - 1) / POSES_PER_BLOCK;
    hipLaunchKernelGGL(pose_aa_kernel, dim3(grid), dim3(POSES_PER_BLOCK), 0,
                       stream, glb, ori, out, npose);
}